// baselineGATv2_35553739276821
// MI455X (gfx1250) — compile-verified
//
#include <hip/hip_runtime.h>

#define N_NODES 50000
#define N_EDGES 800000
#define E_TOTAL (N_EDGES + N_NODES)   // edges + self loops = 850000
#define HEADS 2
#define CH 64
#define HC (HEADS * CH)               // 128
#define LAYERS 3
#define OUTC 128
#define NGRAPH 512

typedef __attribute__((ext_vector_type(2))) float v2f;
typedef __attribute__((ext_vector_type(8))) float v8f;

// ---------------------------------------------------------------------------
// Generic zero fill
// ---------------------------------------------------------------------------
__global__ void zero_f32(float* __restrict__ p, long long n) {
    long long i = (long long)blockIdx.x * blockDim.x + threadIdx.x;
    if (i < n) p[i] = 0.0f;
}

// ---------------------------------------------------------------------------
// Embedding gather: h[n][c] = embed[x[n]][c]
// ---------------------------------------------------------------------------
__global__ void gather_embed(const int* __restrict__ x,
                             const float* __restrict__ embed,
                             float* __restrict__ h) {
    int idx = blockIdx.x * blockDim.x + threadIdx.x;
    if (idx >= N_NODES * CH) return;
    int n = idx >> 6, c = idx & 63;
    h[idx] = embed[(size_t)x[n] * CH + c];
}

// ---------------------------------------------------------------------------
// WMMA GEMM, K fixed at 64 (fp32): D[M,NCOLS] = A[M,64] @ B[64,NCOLS] + bias
// One wave computes a 16x32 output strip (two 16x16 tiles sharing A fetches):
// 16 K-steps x 2 tiles = 32 x v_wmma_f32_16x16x4_f32 per wave.
// NCOLS is a template constant so every access is base + immediate offset.
// ---------------------------------------------------------------------------
template <int NCOLS, int RELU>
__global__ __launch_bounds__(256)
void wmma_gemm64(const float* __restrict__ A, const float* __restrict__ B,
                 const float* __restrict__ bias, float* __restrict__ D, int M) {
    const int lane  = threadIdx.x & 31;
    const int wave  = threadIdx.x >> 5;
    const int tileM = blockIdx.x * 8 + wave;
    if (tileM * 16 >= M) return;                 // wave-uniform: EXEC stays all-1s
    const int col0 = (blockIdx.y << 5) + (lane & 15);   // first 16-col tile
    const int hi   = lane >> 4;                          // half-wave selector

    // A 16x4 fp32 layout: lanes 0-15 -> K=kb+0,kb+1 ; lanes 16-31 -> K=kb+2,kb+3
    const float* Abase  = A + (size_t)((tileM << 4) + (lane & 15)) * CH + 2 * hi;
    // B 4x16 fp32 layout: VGPR0 = rows kb+2*hi, VGPR1 = rows kb+2*hi+1
    const float* B0base = B + (size_t)(2 * hi) * NCOLS + col0;
    const float* B1base = B0base + NCOLS;

    v8f acc0 = {}, acc1 = {};
#pragma unroll
    for (int kb = 0; kb < CH; kb += 4) {
        v2f a = *reinterpret_cast<const v2f*>(Abase + kb);   // global_load_b64
        v2f b0, b1;
        b0.x = B0base[kb * NCOLS];
        b0.y = B1base[kb * NCOLS];
        b1.x = B0base[kb * NCOLS + 16];
        b1.y = B1base[kb * NCOLS + 16];
        acc0 = __builtin_amdgcn_wmma_f32_16x16x4_f32(
            false, a, false, b0, (short)0, acc0, false, false);
        acc1 = __builtin_amdgcn_wmma_f32_16x16x4_f32(
            false, a, false, b1, (short)0, acc1, false, false);
    }

    const float bv0 = bias[col0];
    const float bv1 = bias[col0 + 16];
    // C/D layout: VGPR v holds rows v (lanes 0-15) and v+8 (lanes 16-31)
    float* Dbase = D + ((size_t)(tileM << 4) + 8 * hi) * NCOLS + col0;
#pragma unroll
    for (int v = 0; v < 8; ++v) {
        float v0 = acc0[v] + bv0;
        float v1 = acc1[v] + bv1;
        if (RELU) { v0 = fmaxf(v0, 0.0f); v1 = fmaxf(v1, 0.0f); }
        Dbase[v * NCOLS]      = v0;
        Dbase[v * NCOLS + 16] = v1;
    }
}

// ---------------------------------------------------------------------------
// Per-edge GATv2 score + segment max (one wave per edge).
// lanes 0-15 handle head 0, lanes 16-31 head 1; 4 channels per lane (b128 load)
// ---------------------------------------------------------------------------
__global__ __launch_bounds__(256)
void edge_score_max(const int* __restrict__ ei,
                    const float* __restrict__ xl, const float* __restrict__ xr,
                    const float* __restrict__ att,
                    float* __restrict__ score, unsigned* __restrict__ mkey) {
    int e = blockIdx.x * 8 + (threadIdx.x >> 5);
    if (e >= E_TOTAL) return;
    int lane = threadIdx.x & 31;
    int src, dst;
    if (e < N_EDGES) { src = ei[e]; dst = ei[N_EDGES + e]; }
    else             { src = dst = e - N_EDGES; }

    const int h  = lane >> 4;
    const int c0 = (lane & 15) << 2;
    const float4 vl = *reinterpret_cast<const float4*>(xl + (size_t)src * HC + h * CH + c0);
    const float4 vr = *reinterpret_cast<const float4*>(xr + (size_t)dst * HC + h * CH + c0);
    const float4 va = *reinterpret_cast<const float4*>(att + h * CH + c0);

    float s = 0.0f;
    {
        float t;
        t = vl.x + vr.x; t = (t > 0.0f) ? t : 0.2f * t; s += t * va.x;
        t = vl.y + vr.y; t = (t > 0.0f) ? t : 0.2f * t; s += t * va.y;
        t = vl.z + vr.z; t = (t > 0.0f) ? t : 0.2f * t; s += t * va.z;
        t = vl.w + vr.w; t = (t > 0.0f) ? t : 0.2f * t; s += t * va.w;
    }
    // reduce within each 16-lane head group (wave32)
#pragma unroll
    for (int off = 1; off < 16; off <<= 1) s += __shfl_xor(s, off, 32);

    if ((lane & 15) == 0) {
        score[(size_t)e * HEADS + h] = s;
        unsigned b = __float_as_uint(s);
        unsigned key = (b & 0x80000000u) ? ~b : (b | 0x80000000u); // order-preserving
        atomicMax(mkey + (size_t)dst * HEADS + h, key);
    }
}

// ---------------------------------------------------------------------------
// p = exp(score - max[dst]); denom[dst] += p   (p stored in place of score)
// ---------------------------------------------------------------------------
__global__ void edge_exp_denom(const int* __restrict__ ei,
                               float* __restrict__ score,
                               const unsigned* __restrict__ mkey,
                               float* __restrict__ denom) {
    int idx = blockIdx.x * blockDim.x + threadIdx.x;     // over E_TOTAL * HEADS
    if (idx >= E_TOTAL * HEADS) return;
    int e = idx >> 1, h = idx & 1;
    int dst = (e < N_EDGES) ? ei[N_EDGES + e] : e - N_EDGES;
    unsigned key = mkey[(size_t)dst * HEADS + h];
    unsigned b = (key & 0x80000000u) ? (key ^ 0x80000000u) : ~key;
    float m = __uint_as_float(b);
    float p = __expf(score[idx] - m);
    score[idx] = p;
    atomicAdd(denom + (size_t)dst * HEADS + h, p);
}

// ---------------------------------------------------------------------------
// agg[dst] += alpha * xl[src]  (one wave per edge, 4 channels per lane)
// ---------------------------------------------------------------------------
__global__ __launch_bounds__(256)
void edge_aggregate(const int* __restrict__ ei, const float* __restrict__ xl,
                    const float* __restrict__ p, const float* __restrict__ denom,
                    float* __restrict__ agg) {
    int e = blockIdx.x * 8 + (threadIdx.x >> 5);
    if (e >= E_TOTAL) return;
    int lane = threadIdx.x & 31;
    int src, dst;
    if (e < N_EDGES) { src = ei[e]; dst = ei[N_EDGES + e]; }
    else             { src = dst = e - N_EDGES; }

    const int h  = lane >> 4;
    const int c0 = (lane & 15) << 2;
    const float alpha = p[(size_t)e * HEADS + h] / denom[(size_t)dst * HEADS + h];
    const float4 vl = *reinterpret_cast<const float4*>(xl + (size_t)src * HC + h * CH + c0);
    float* pd = agg + (size_t)dst * HC + h * CH + c0;
    atomicAdd(pd + 0, vl.x * alpha);
    atomicAdd(pd + 1, vl.y * alpha);
    atomicAdd(pd + 2, vl.z * alpha);
    atomicAdd(pd + 3, vl.w * alpha);
}

// ---------------------------------------------------------------------------
// head mean + layer bias: h[n][c] = 0.5*(agg[n][0][c]+agg[n][1][c]) + bias[c]
// ---------------------------------------------------------------------------
__global__ void head_mean_bias(const float* __restrict__ agg,
                               const float* __restrict__ bias,
                               float* __restrict__ hout) {
    int idx = blockIdx.x * blockDim.x + threadIdx.x;
    if (idx >= N_NODES * CH) return;
    int n = idx >> 6, c = idx & 63;
    hout[idx] = 0.5f * (agg[(size_t)n * HC + c] + agg[(size_t)n * HC + CH + c]) + bias[c];
}

// ---------------------------------------------------------------------------
// Graph readout: d_out[batch[n]] += out[n]
// ---------------------------------------------------------------------------
__global__ void graph_sum(const float* __restrict__ out,
                          const int* __restrict__ batch,
                          float* __restrict__ gout) {
    int idx = blockIdx.x * blockDim.x + threadIdx.x;
    if (idx >= N_NODES * OUTC) return;
    int n = idx >> 7, c = idx & 127;
    atomicAdd(gout + (size_t)batch[n] * OUTC + c, out[idx]);
}

// ---------------------------------------------------------------------------
extern "C" void kernel_launch(void* const* d_in, const int* in_sizes, int n_in,
                              void* d_out, int out_size, void* d_ws, size_t ws_size,
                              hipStream_t stream) {
    const int*   x        = (const int*)d_in[0];
    const int*   ei       = (const int*)d_in[1];
    const int*   batch    = (const int*)d_in[2];
    const float* embed    = (const float*)d_in[4];
    const float* lin_W    = (const float*)d_in[5];
    const float* lin_b    = (const float*)d_in[6];
    const float* gat_Wl   = (const float*)d_in[7];
    const float* gat_bl   = (const float*)d_in[8];
    const float* gat_Wr   = (const float*)d_in[9];
    const float* gat_br   = (const float*)d_in[10];
    const float* gat_att  = (const float*)d_in[11];
    const float* gat_bias = (const float*)d_in[12];
    const float* ro_W     = (const float*)d_in[13];
    const float* ro_b     = (const float*)d_in[14];
    float* gout = (float*)d_out;

    // workspace layout (floats)
    float* ws = (float*)d_ws;
    size_t off = 0;
    float*    h     = ws + off; off += (size_t)N_NODES * CH;
    float*    hlin  = ws + off; off += (size_t)N_NODES * CH;
    float*    xl    = ws + off; off += (size_t)N_NODES * HC;
    float*    xr    = ws + off; off += (size_t)N_NODES * HC;
    float*    sc    = ws + off; off += (size_t)E_TOTAL * HEADS;   // score, then p in place
    unsigned* mkey  = (unsigned*)(ws + off); off += (size_t)N_NODES * HEADS;
    float*    denom = ws + off; off += (size_t)N_NODES * HEADS;
    float*    agg   = ws + off; off += (size_t)N_NODES * HC;
    float*    outb  = ws + off; off += (size_t)N_NODES * OUTC;

    const int mTiles = N_NODES / 16;                       // 3125
    dim3 blk256(256);
    dim3 gemmGrd64((mTiles + 7) / 8, CH / 32);             // NCOLS = 64  -> y=2
    dim3 gemmGrd128((mTiles + 7) / 8, HC / 32);            // NCOLS = 128 -> y=4
    const int nc64  = N_NODES * CH;
    const int nc128 = N_NODES * OUTC;
    const long long zeroSpan = (long long)N_NODES * (HEADS + HEADS + HC); // mkey+denom+agg

    gather_embed<<<(nc64 + 255) / 256, blk256, 0, stream>>>(x, embed, h);

    for (int l = 0; l < LAYERS; ++l) {
        // lin + ReLU : hlin = relu(h @ lin_W[l] + lin_b[l])
        wmma_gemm64<CH, 1><<<gemmGrd64, blk256, 0, stream>>>(
            h, lin_W + (size_t)l * CH * CH, lin_b + (size_t)l * CH, hlin, N_NODES);
        // xl = hlin @ Wl + bl ; xr = hlin @ Wr + br
        wmma_gemm64<HC, 0><<<gemmGrd128, blk256, 0, stream>>>(
            hlin, gat_Wl + (size_t)l * CH * HC, gat_bl + (size_t)l * HC, xl, N_NODES);
        wmma_gemm64<HC, 0><<<gemmGrd128, blk256, 0, stream>>>(
            hlin, gat_Wr + (size_t)l * CH * HC, gat_br + (size_t)l * HC, xr, N_NODES);
        // clear mkey / denom / agg (contiguous in ws)
        zero_f32<<<(unsigned)((zeroSpan + 255) / 256), blk256, 0, stream>>>(
            (float*)mkey, zeroSpan);
        // scores + segment max
        edge_score_max<<<E_TOTAL / 8, blk256, 0, stream>>>(
            ei, xl, xr, gat_att + (size_t)l * HEADS * CH, sc, mkey);
        // exp + denom
        edge_exp_denom<<<(E_TOTAL * HEADS + 255) / 256, blk256, 0, stream>>>(
            ei, sc, mkey, denom);
        // weighted aggregation
        edge_aggregate<<<E_TOTAL / 8, blk256, 0, stream>>>(ei, xl, sc, denom, agg);
        // head mean + bias -> h (next layer input)
        head_mean_bias<<<(nc64 + 255) / 256, blk256, 0, stream>>>(
            agg, gat_bias + (size_t)l * CH, h);
    }

    // readout: outb = h @ readout_W + readout_b
    wmma_gemm64<OUTC, 0><<<gemmGrd128, blk256, 0, stream>>>(
        h, ro_W, ro_b, outb, N_NODES);

    // d_out must be zeroed every call (harness does not re-poison between replays)
    zero_f32<<<(NGRAPH * OUTC + 255) / 256, blk256, 0, stream>>>(
        gout, (long long)NGRAPH * OUTC);
    graph_sum<<<(nc128 + 255) / 256, blk256, 0, stream>>>(outb, batch, gout);
}